// GraphClassifierWhole_12773232739013
// MI455X (gfx1250) — compile-verified
//
#include <hip/hip_runtime.h>

typedef float v2f __attribute__((ext_vector_type(2)));
typedef float v8f __attribute__((ext_vector_type(8)));

#define PSD    256     // L*D
#define DDIM   64
#define RELD   32
#define INP    32
#define GDIM   864     // 2*PSD + RELD + PSD + 2*INP
#define MDDIM  320     // PSD + D
#define H1DIM  64
#define H2DIM  32
#define SSET   32

// g_rep column offsets (concat order from reference); all 16B-aligned (x4)
#define OFF_HEAD 0
#define OFF_TAIL 256
#define OFF_REL  512
#define OFF_MID  544
#define OFF_HF   800
#define OFF_TF   832

// LDS strides: multiples of 4 floats (16B, so b128 ok) AND row*stride mod 64
// distinct over 16 rows (conflict-free A-fragment column reads).
#define SA_STRIDE  868   // 868 mod 64 = 36, 16 rows -> 16 distinct banks
#define SD_STRIDE  68    // 68 mod 64 = 4,  16 rows -> 16 distinct banks
#define SH1_STRIDE 65    // scalar access only; odd -> conflict-free
#define SH2_STRIDE 33    // scalar access only

__global__ __launch_bounds__(256)
void gcw_fused_kernel(const float* __restrict__ node_repr,   // [N,256]
                      const float* __restrict__ node_feat,   // [N,32]
                      const int*   __restrict__ head_idx,    // [B]
                      const int*   __restrict__ rel_idx,     // [B]
                      const int*   __restrict__ tail_idx,    // [B]
                      const int*   __restrict__ dist,        // [B]
                      const int*   __restrict__ inter,       // [B,32]
                      const float* __restrict__ rel_emb,     // [200,32]
                      const float* __restrict__ dist_emb,    // [11,64]
                      const float* __restrict__ head_W,      // [320,32]
                      const float* __restrict__ head_b,      // [32]
                      const float* __restrict__ tail_W,      // [320,32]
                      const float* __restrict__ tail_b,      // [32]
                      const float* __restrict__ W1,          // [864,64]
                      const float* __restrict__ b1,          // [64]
                      const float* __restrict__ W2,          // [64,32]
                      const float* __restrict__ b2,          // [32]
                      const float* __restrict__ W3,          // [32,1]
                      const float* __restrict__ b3,          // [1]
                      float* __restrict__ out0,              // [B]
                      float* __restrict__ out_hp,            // [B,32]
                      float* __restrict__ out_tp,            // [B,32]
                      float* __restrict__ out_hf,            // [B,32]
                      float* __restrict__ out_tf,            // [B,32]
                      int Btot)
{
    __shared__ float smem[16 * SA_STRIDE + 16 * SD_STRIDE + 16 * SH1_STRIDE];
    float* sA  = smem;                       // 16 x 864 g_rep tile (padded)
    float* sD  = smem + 16 * SA_STRIDE;      // 16 x 64 dist tile (later: h2 tile)
    float* sH1 = sD   + 16 * SD_STRIDE;      // 16 x 64 relu(W1) tile
    float* sH2 = sD;                         // alias: reused after second barrier

    const int t        = threadIdx.x;        // 0..255
    const int linkBase = blockIdx.x * 16;
    const bool full    = (linkBase + 16 <= Btot);  // uniform fast-path flag

    const int g = t >> 6;                    // 64-lane group 0..3
    const int j = t & 63;                    // lane in group

    // ---------------- Phase 1a: embedding gathers into LDS (b128-wide) ----
    for (int li = 0; li < 16; ++li) {
        int link = linkBase + li;
        if (link >= Btot) link = Btot - 1;
        if (g == 0) {                        // head row: 64 lanes x float4 = 1KB
            const int h = head_idx[link];
            const float4 v = ((const float4*)(node_repr + (size_t)h * PSD))[j];
            *(float4*)&sA[li * SA_STRIDE + OFF_HEAD + 4 * j] = v;
        } else if (g == 1) {                 // tail row
            const int tl = tail_idx[link];
            const float4 v = ((const float4*)(node_repr + (size_t)tl * PSD))[j];
            *(float4*)&sA[li * SA_STRIDE + OFF_TAIL + 4 * j] = v;
        } else if (g == 2) {                 // all the small pieces
            if (j < 16) {                    // dist: 16 x float4
                const float4 v = ((const float4*)(dist_emb + dist[link] * DDIM))[j];
                *(float4*)&sD[li * SD_STRIDE + 4 * j] = v;
            } else if (j < 24) {             // rel: 8 x float4
                const float4 v = ((const float4*)(rel_emb + rel_idx[link] * RELD))[j - 16];
                *(float4*)&sA[li * SA_STRIDE + OFF_REL + 4 * (j - 16)] = v;
            } else if (j < 32) {             // head feat: 8 x float4 (LDS + out)
                const int h  = head_idx[link];
                const int jj = j - 24;
                const float4 v = ((const float4*)(node_feat + (size_t)h * INP))[jj];
                *(float4*)&sA[li * SA_STRIDE + OFF_HF + 4 * jj] = v;
                if (linkBase + li < Btot)
                    ((float4*)(out_hf + (size_t)link * INP))[jj] = v;
            } else if (j < 40) {             // tail feat: 8 x float4 (LDS + out)
                const int tl = tail_idx[link];
                const int jj = j - 32;
                const float4 v = ((const float4*)(node_feat + (size_t)tl * INP))[jj];
                *(float4*)&sA[li * SA_STRIDE + OFF_TF + 4 * jj] = v;
                if (linkBase + li < Btot)
                    ((float4*)(out_tf + (size_t)link * INP))[jj] = v;
            }
        }
    }

    // ---------------- Phase 1b: masked mean over intersection set ---------
    // 4 links in flight (one per 64-lane group), float4 per lane: each row
    // read is a fully coalesced 1KB b128 burst.
    {
        const int c4 = 4 * j;                // column base 0..252
        for (int li = g; li < 16; li += 4) {
            int link = linkBase + li;
            if (link >= Btot) link = Btot - 1;
            float ax = 0.f, ay = 0.f, az = 0.f, aw = 0.f;
            int cnt = 0;
            #pragma unroll 4
            for (int s = 0; s < SSET; ++s) {
                const int v  = inter[link * SSET + s];      // uniform -> scalar load
                const int ia = v < 0 ? -v : v;
                const float sg = (v + 1 > 0) ? 1.0f : ((v + 1 == 0) ? 0.0f : -1.0f);
                cnt += (v != -1) ? 1 : 0;
                const float4 r = *(const float4*)(node_repr + (size_t)ia * PSD + c4);
                ax += sg * r.x; ay += sg * r.y; az += sg * r.z; aw += sg * r.w;
            }
            if (cnt < 1) cnt = 1;
            const float fc = (float)cnt;
            float4 m;
            m.x = ax / fc; m.y = ay / fc; m.z = az / fc; m.w = aw / fc;
            *(float4*)&sA[li * SA_STRIDE + OFF_MID + c4] = m;
        }
    }
    __syncthreads();

    // ---------------- Phase 2: WMMA GEMMs (wave-uniform control flow) -----
    const int wv   = t >> 5;       // wave id 0..7
    const int ln   = t & 31;       // lane in wave
    const int half = ln >> 4;      // 0: K0/K1 side, 1: K2/K3 side
    const int nl   = ln & 15;      // A row / B-C column within tile

    if (wv < 4) {
        // g_rep[16x864] @ W1[864x64] -> relu -> sH1 ; wave wv owns N-tile wv*16
        const int n0 = wv * 16;
        v8f acc = {};
        #pragma unroll 4
        for (int k0 = 0; k0 < GDIM; k0 += 4) {
            const int ka = k0 + 2 * half;
            v2f a, b;
            a[0] = sA[nl * SA_STRIDE + ka];
            a[1] = sA[nl * SA_STRIDE + ka + 1];
            b[0] = W1[ka * H1DIM + n0 + nl];
            b[1] = W1[(ka + 1) * H1DIM + n0 + nl];
            acc = __builtin_amdgcn_wmma_f32_16x16x4_f32(
                false, a, false, b, (short)0, acc, false, false);
        }
        const float bias = b1[n0 + nl];
        #pragma unroll
        for (int jj = 0; jj < 8; ++jj) {
            float v = acc[jj] + bias;
            v = v > 0.0f ? v : 0.0f;
            sH1[(jj + 8 * half) * SH1_STRIDE + n0 + nl] = v;
        }
    } else {
        // md[16x320] @ {head_W|tail_W}[320x32] -> head_pred / tail_pred
        const float* W    = (wv < 6) ? head_W : tail_W;
        const float* bb   = (wv < 6) ? head_b : tail_b;
        float*       outp = (wv < 6) ? out_hp : out_tp;
        const int n0 = (wv & 1) * 16;
        v8f acc = {};
        #pragma unroll 4
        for (int k0 = 0; k0 < MDDIM; k0 += 4) {
            const int ka = k0 + 2 * half;
            v2f a, b;
            if (k0 < PSD) {            // mid_repr region of g_rep
                a[0] = sA[nl * SA_STRIDE + OFF_MID + ka];
                a[1] = sA[nl * SA_STRIDE + OFF_MID + ka + 1];
            } else {                   // dist_repr tile
                a[0] = sD[nl * SD_STRIDE + (ka - PSD)];
                a[1] = sD[nl * SD_STRIDE + (ka - PSD) + 1];
            }
            b[0] = W[ka * INP + n0 + nl];
            b[1] = W[(ka + 1) * INP + n0 + nl];
            acc = __builtin_amdgcn_wmma_f32_16x16x4_f32(
                false, a, false, b, (short)0, acc, false, false);
        }
        const float bias = bb[n0 + nl];
        if (full) {                    // uniform: straight-line stores
            #pragma unroll
            for (int jj = 0; jj < 8; ++jj) {
                const int li = jj + 8 * half;
                outp[(size_t)(linkBase + li) * INP + n0 + nl] = acc[jj] + bias;
            }
        } else {
            #pragma unroll
            for (int jj = 0; jj < 8; ++jj) {
                const int li = jj + 8 * half;
                if (linkBase + li < Btot)
                    outp[(size_t)(linkBase + li) * INP + n0 + nl] = acc[jj] + bias;
            }
        }
    }
    __syncthreads();   // sH1 complete; sD no longer read -> safe to alias as sH2

    if (wv < 2) {
        // h1[16x64] @ W2[64x32] -> relu -> sH2
        const int n0 = wv * 16;
        v8f acc = {};
        #pragma unroll
        for (int k0 = 0; k0 < H1DIM; k0 += 4) {
            const int ka = k0 + 2 * half;
            v2f a, b;
            a[0] = sH1[nl * SH1_STRIDE + ka];
            a[1] = sH1[nl * SH1_STRIDE + ka + 1];
            b[0] = W2[ka * H2DIM + n0 + nl];
            b[1] = W2[(ka + 1) * H2DIM + n0 + nl];
            acc = __builtin_amdgcn_wmma_f32_16x16x4_f32(
                false, a, false, b, (short)0, acc, false, false);
        }
        const float bias = b2[n0 + nl];
        #pragma unroll
        for (int jj = 0; jj < 8; ++jj) {
            float v = acc[jj] + bias;
            v = v > 0.0f ? v : 0.0f;
            sH2[(jj + 8 * half) * SH2_STRIDE + n0 + nl] = v;
        }
    }
    __syncthreads();

    // h2[16x32] @ W3[32x1] + b3
    if (t < 16) {
        const int link = linkBase + t;
        float s = b3[0];
        #pragma unroll
        for (int jj = 0; jj < H2DIM; ++jj)
            s += sH2[t * SH2_STRIDE + jj] * W3[jj];
        if (link < Btot) out0[link] = s;
    }
}

extern "C" void kernel_launch(void* const* d_in, const int* in_sizes, int n_in,
                              void* d_out, int out_size, void* d_ws, size_t ws_size,
                              hipStream_t stream) {
    (void)n_in; (void)out_size; (void)d_ws; (void)ws_size;
    const float* node_repr = (const float*)d_in[0];
    const float* node_feat = (const float*)d_in[1];
    const int*   head_idx  = (const int*)  d_in[2];
    const int*   rel_idx   = (const int*)  d_in[3];
    const int*   tail_idx  = (const int*)  d_in[4];
    const int*   dist      = (const int*)  d_in[5];
    const int*   inter     = (const int*)  d_in[6];
    const float* rel_emb   = (const float*)d_in[7];
    const float* dist_emb  = (const float*)d_in[8];
    const float* head_W    = (const float*)d_in[9];
    const float* head_b    = (const float*)d_in[10];
    const float* tail_W    = (const float*)d_in[11];
    const float* tail_b    = (const float*)d_in[12];
    const float* W1        = (const float*)d_in[13];
    const float* b1        = (const float*)d_in[14];
    const float* W2        = (const float*)d_in[15];
    const float* b2        = (const float*)d_in[16];
    const float* W3        = (const float*)d_in[17];
    const float* b3        = (const float*)d_in[18];

    const int B = in_sizes[2];             // number of links
    float* out = (float*)d_out;
    float* out0    = out;                  // [B,1]
    float* out_hp  = out + (size_t)B;      // [B,32]
    float* out_tp  = out + (size_t)B * 33; // [B,32]
    float* out_hf  = out + (size_t)B * 65; // [B,32]
    float* out_tf  = out + (size_t)B * 97; // [B,32]

    const int nBlocks = (B + 15) / 16;
    gcw_fused_kernel<<<nBlocks, 256, 0, stream>>>(
        node_repr, node_feat, head_idx, rel_idx, tail_idx, dist, inter,
        rel_emb, dist_emb, head_W, head_b, tail_W, tail_b,
        W1, b1, W2, b2, W3, b3,
        out0, out_hp, out_tp, out_hf, out_tf, B);
}